// CoverTreeLoss_90589450207920
// MI455X (gfx1250) — compile-verified
//
#include <hip/hip_runtime.h>

#define C_CLS 10000
#define D_    512
#define B_    4096
#define P_    12

#define BM 128
#define BN 128
#define BK 32
#define KP 18   // K-pair row stride (in v2f units) = 36 floats; conflict-free (9 coprime 16)

typedef __attribute__((ext_vector_type(2))) float v2f;
typedef __attribute__((ext_vector_type(8))) float v8f;

// ---------------------------------------------------------------------------
// Kernel 1: added_weights[c,:] = sum_p mask[c,p] * weights[path_idx[c,p], :]
// mask is a prefix mask (arange(P) < depth) -> early exit on first zero.
// ---------------------------------------------------------------------------
__global__ __launch_bounds__(256) void build_aw_kernel(
    const float* __restrict__ w, const int* __restrict__ pidx,
    const float* __restrict__ pmask, float* __restrict__ aw) {
  const int c = blockIdx.x;
  const int t = threadIdx.x;
  float a0 = 0.f, a1 = 0.f;
#pragma unroll 1
  for (int p = 0; p < P_; ++p) {
    const float m = pmask[c * P_ + p];
    if (m == 0.f) break;  // prefix mask
    const int r = pidx[c * P_ + p];
    const float* row = w + (size_t)r * D_;
    a0 = fmaf(m, row[t], a0);
    a1 = fmaf(m, row[t + 256], a1);
  }
  aw[(size_t)c * D_ + t] = a0;
  aw[(size_t)c * D_ + t + 256] = a1;
}

// ---------------------------------------------------------------------------
// Kernel 2: logits[B,C] = x[B,D] @ added_weights[C,D]^T via V_WMMA_F32_16X16X4_F32
// K-pair LDS layout: each fragment = one naturally aligned ds_load_b64 into a
// consecutive VGPR pair (no v_dual_mov re-pairing before the WMMA).
// ---------------------------------------------------------------------------
__global__ __launch_bounds__(256) void gemm_wmma_kernel(
    const float* __restrict__ X, const float* __restrict__ AW,
    float* __restrict__ OUT /* logits, 4-byte aligned only */) {
  __shared__ v2f Ap[BM * KP];  // [m][kpair]
  __shared__ v2f Bp[BN * KP];  // [n][kpair]

  const int tid  = threadIdx.x;
  const int lane = tid & 31;
  const int wave = tid >> 5;  // 0..7
  const int wm   = wave >> 1; // 0..3 -> 32-row band
  const int wn   = wave & 1;  // 0..1 -> 64-col band
  const int half = lane >> 4; // selects K pair 0/1 within the 4-wide K step
  const int l16  = lane & 15;

  const int mBase = blockIdx.y * BM;
  const int nBase = blockIdx.x * BN;

  // staging: 2 threads per row, each covers 16 floats (4 float4 chunks)
  const int srow = tid >> 1;            // 0..127
  const int sk0  = (tid & 1) * 16;      // 0 or 16
  const int gnS  = nBase + srow;        // B staging row (guard vs C)
  const float* aRow = X + (size_t)(mBase + srow) * D_;
  const float* bRow = AW + (size_t)(gnS < C_CLS ? gnS : 0) * D_;
  const bool bValid = (gnS < C_CLS);

  v8f acc[2][4] = {};

  for (int k0 = 0; k0 < D_; k0 += BK) {
    if (k0 + BK < D_) {  // prefetch next chunk (global_prefetch_b8)
      __builtin_prefetch(aRow + k0 + BK + sk0, 0, 3);
      __builtin_prefetch(bRow + k0 + BK + sk0, 0, 3);
    }
    // stage A: float4 global load (16B-aligned) -> one ds_store_b128
#pragma unroll
    for (int j = 0; j < 4; ++j) {
      const int k = sk0 + j * 4;
      const float4 v = *(const float4*)(aRow + k0 + k);
      *(float4*)(&Ap[srow * KP + (k >> 1)]) = v;
    }
    // stage B: 4 scalar global loads (dst only 4B-aligned) -> one ds_store_b128
#pragma unroll
    for (int j = 0; j < 4; ++j) {
      const int k = sk0 + j * 4;
      float4 v;
      if (bValid) {
        v.x = bRow[k0 + k + 0];
        v.y = bRow[k0 + k + 1];
        v.z = bRow[k0 + k + 2];
        v.w = bRow[k0 + k + 3];
      } else {
        v.x = v.y = v.z = v.w = 0.f;
      }
      *(float4*)(&Bp[srow * KP + (k >> 1)]) = v;
    }
    __syncthreads();

#pragma unroll
    for (int kk = 0; kk < BK; kk += 4) {
      const int kp = (kk >> 1) + half;  // K-pair index for this lane half
      v2f a[2], b[4];
#pragma unroll
      for (int mi = 0; mi < 2; ++mi)
        a[mi] = Ap[(wm * 32 + mi * 16 + l16) * KP + kp];  // ds_load_b64
#pragma unroll
      for (int ni = 0; ni < 4; ++ni)
        b[ni] = Bp[(wn * 64 + ni * 16 + l16) * KP + kp];  // ds_load_b64
#pragma unroll
      for (int mi = 0; mi < 2; ++mi)
#pragma unroll
        for (int ni = 0; ni < 4; ++ni)
          acc[mi][ni] = __builtin_amdgcn_wmma_f32_16x16x4_f32(
              false, a[mi], false, b[ni], (short)0, acc[mi][ni], false, false);
    }
    __syncthreads();
  }

  // D layout: VGPR r -> row (r + 8*half), col l16. Scalar stores (dst 4B-aligned).
#pragma unroll
  for (int mi = 0; mi < 2; ++mi) {
#pragma unroll
    for (int ni = 0; ni < 4; ++ni) {
      const int gn = nBase + wn * 64 + ni * 16 + l16;
      if (gn < C_CLS) {
#pragma unroll
        for (int r = 0; r < 8; ++r) {
          const int gm = mBase + wm * 32 + mi * 16 + r + 8 * half;
          OUT[(size_t)gm * C_CLS + gn] = acc[mi][ni][r];
        }
      }
    }
  }
}

// ---------------------------------------------------------------------------
// Kernel 3: per-row online logsumexp -> nll[row] = (max + log sumexp) - logit[y]
// ---------------------------------------------------------------------------
__global__ __launch_bounds__(256) void row_nll_kernel(
    const float* __restrict__ logits, const int* __restrict__ y,
    float* __restrict__ rownll) {
  __shared__ float sm[256];
  __shared__ float ss[256];
  const int row = blockIdx.x;
  const int t = threadIdx.x;
  const float* p = logits + (size_t)row * C_CLS;
  float m = -3.402823466e38f, s = 0.f;
  for (int i = t; i < C_CLS; i += 256) {
    const float v = p[i];
    const float nm = fmaxf(m, v);
    s = s * __expf(m - nm) + __expf(v - nm);
    m = nm;
  }
  sm[t] = m; ss[t] = s;
  __syncthreads();
  for (int off = 128; off > 0; off >>= 1) {
    if (t < off) {
      const float m2 = sm[t + off], s2 = ss[t + off];
      const float M = fmaxf(sm[t], m2);
      ss[t] = ss[t] * __expf(sm[t] - M) + s2 * __expf(m2 - M);
      sm[t] = M;
    }
    __syncthreads();
  }
  if (t == 0) rownll[row] = (sm[0] + __logf(ss[0])) - p[y[row]];
}

// ---------------------------------------------------------------------------
// Kernel 4: loss = mean(rownll)
// ---------------------------------------------------------------------------
__global__ __launch_bounds__(256) void loss_reduce_kernel(
    const float* __restrict__ rownll, float* __restrict__ out) {
  __shared__ float sb[256];
  const int t = threadIdx.x;
  float a = 0.f;
  for (int i = t; i < B_; i += 256) a += rownll[i];
  sb[t] = a;
  __syncthreads();
  for (int off = 128; off > 0; off >>= 1) {
    if (t < off) sb[t] += sb[t + off];
    __syncthreads();
  }
  if (t == 0) out[0] = sb[0] / (float)B_;
}

extern "C" void kernel_launch(void* const* d_in, const int* in_sizes, int n_in,
                              void* d_out, int out_size, void* d_ws, size_t ws_size,
                              hipStream_t stream) {
  const float* w     = (const float*)d_in[0];  // [LENGTH, D]
  const float* x     = (const float*)d_in[1];  // [B, D]
  const int*   y     = (const int*)d_in[2];    // [B]
  const int*   pidx  = (const int*)d_in[3];    // [C, P]
  const float* pmask = (const float*)d_in[4];  // [C, P]

  float* out    = (float*)d_out;
  float* logits = out + 1;                          // [B, C]
  float* aw     = out + 1 + (size_t)B_ * C_CLS;     // [C, D]
  float* rownll = (float*)d_ws;                     // [B] scratch

  build_aw_kernel<<<C_CLS, 256, 0, stream>>>(w, pidx, pmask, aw);

  dim3 grid((C_CLS + BN - 1) / BN, B_ / BM);  // 79 x 32
  gemm_wmma_kernel<<<grid, 256, 0, stream>>>(x, aw, logits);

  row_nll_kernel<<<B_, 256, 0, stream>>>(logits, y, rownll);
  loss_reduce_kernel<<<1, 256, 0, stream>>>(rownll, out);
}